// SSC_Model_separable_10179072492157
// MI455X (gfx1250) — compile-verified
//
#include <hip/hip_runtime.h>
#include <hip/hip_bf16.h>
#include <math.h>

#define LAMBD  0.2f
#define T_MAX  1000
#define N_SAMP 1024
#define NDIM   64
#define MDIM   256
#define NB     32      // samples per ISTA workgroup
#define SMAX   96      // max support size handled by refinement

typedef __attribute__((ext_vector_type(2))) float v2f;
typedef __attribute__((ext_vector_type(8))) float v8f;

__device__ __forceinline__ v8f wmma4(v2f a, v2f b, v8f c) {
  // D = A(16x4 f32) * B(4x16 f32) + C(16x16 f32)
  return __builtin_amdgcn_wmma_f32_16x16x4_f32(
      /*neg_a=*/false, a, /*neg_b=*/false, b,
      /*c_mod=*/(short)0, c, /*reuse_a=*/false, /*reuse_b=*/false);
}

// ---------------------------------------------------------------------------
// Kernel 1: G = D^T D (WMMA) + power iteration for eta = 1/sigma_max(D)^2
// ---------------------------------------------------------------------------
__global__ void k_gram_eta(const float* __restrict__ D, float* __restrict__ G,
                           float* __restrict__ etaOut) {
  __shared__ float sD[NDIM * MDIM];   // 64 KB, D row-major (64 x 256)
  __shared__ float sV[MDIM];
  __shared__ float sT[NDIM];
  __shared__ float sRed;
  __shared__ float sLam;

  const int tid = threadIdx.x;        // 256 threads = 8 waves
  for (int i = tid; i < NDIM * MDIM; i += 256) sD[i] = D[i];
  for (int i = tid; i < MDIM; i += 256) sV[i] = 1.f;
  if (tid == 0) { sRed = 0.f; sLam = 1.f; }
  __syncthreads();

  const int wid = tid >> 5, lane = tid & 31, half = lane >> 4, lm = lane & 15;

  // G = D^T D : 256 16x16 output tiles, 32 per wave, K = 64
  for (int t = 0; t < 32; ++t) {
    const int q  = wid + 8 * t;
    const int mb = (q >> 4) * 16;
    const int nb = (q & 15) * 16;
    v8f acc = {0.f, 0.f, 0.f, 0.f, 0.f, 0.f, 0.f, 0.f};
    for (int k = 0; k < NDIM; k += 4) {
      v2f a, b;
      a.x = sD[(k + half * 2 + 0) * MDIM + mb + lm];  // (D^T)[m][k] = D[k][m]
      a.y = sD[(k + half * 2 + 1) * MDIM + mb + lm];
      b.x = sD[(k + half * 2 + 0) * MDIM + nb + lm];  // B[k][n] = D[k][n]
      b.y = sD[(k + half * 2 + 1) * MDIM + nb + lm];
      acc = wmma4(a, b, acc);
    }
#pragma unroll
    for (int r = 0; r < 8; ++r)
      G[(mb + r + 8 * half) * MDIM + nb + lm] = acc[r];
  }
  __syncthreads();

  // power iteration on D^T D using LDS-resident D:  v' = D^T (D v)
  for (int pit = 0; pit < 64; ++pit) {
    if (tid < NDIM) {
      float s = 0.f;
      for (int j = 0; j < MDIM; ++j) s += sD[tid * MDIM + j] * sV[j];
      sT[tid] = s;
    }
    __syncthreads();
    float vn = 0.f;
    for (int k = 0; k < NDIM; ++k) vn += sD[k * MDIM + tid] * sT[k];
    atomicAdd(&sRed, vn * vn);
    __syncthreads();
    const float nrm = sqrtf(sRed);
    sV[tid] = vn / nrm;
    __syncthreads();
    if (tid == 0) { sLam = nrm; sRed = 0.f; }
    __syncthreads();
  }
  if (tid == 0) etaOut[0] = 1.f / sLam;
}

// ---------------------------------------------------------------------------
// Kernel 2: ISTA. One workgroup = 32 samples, 8 waves, D/z/res/x in LDS.
//   phase1: res = D z - x            (8 tiles, 1/wave, 64 WMMA each)
//   phase2: z = shrink(z - eta*D^T res)  (32 tiles, 4/wave, 16 WMMA each)
// ---------------------------------------------------------------------------
__global__ void k_ista(const float* __restrict__ x, const float* __restrict__ D,
                       const float* __restrict__ etaIn, float* __restrict__ zws) {
  __shared__ float sD[NDIM * MDIM];   // 64 KB
  __shared__ float sZ[MDIM * NB];     // 32 KB, z row-major (256 x 32)
  __shared__ float sR[NDIM * NB];     //  8 KB, res (64 x 32)
  __shared__ float sX[NDIM * NB];     //  8 KB, x_t tile (64 x 32)
  __shared__ float sDelta;

  const int tid = threadIdx.x;        // 256 threads
  const int n0  = blockIdx.x * NB;

  for (int i = tid; i < NDIM * MDIM; i += 256) sD[i] = D[i];
  for (int i = tid; i < NDIM * NB; i += 256) {
    const int j = i >> 6, k = i & 63;           // coalesced over k
    sX[k * NB + j] = x[(n0 + j) * NDIM + k];
  }
  for (int i = tid; i < MDIM * NB; i += 256) sZ[i] = 0.f;
  if (tid == 0) sDelta = 0.f;
  const float eta = etaIn[0];
  const float thr = LAMBD * eta;
  __syncthreads();

  const int wid = tid >> 5, lane = tid & 31, half = lane >> 4, lm = lane & 15;
  const int mtR = (wid & 3) * 16;      // phase-1 row-tile base (0..48)
  const int ntR = (wid >> 2) * 16;     // phase-1 col-tile base (0 or 16)

  // z=0 entry makes the first trip produce z0 = shrink(eta*D^T x); then up to
  // T_MAX more trips => 1001 shrink applications, matching the reference.
  for (int it = 0; it <= T_MAX; ++it) {
    { // ---- phase 1: res = D z - x ----
      v8f acc = {0.f, 0.f, 0.f, 0.f, 0.f, 0.f, 0.f, 0.f};
      for (int k = 0; k < MDIM; k += 4) {
        v2f a, b;
        a.x = sD[(mtR + lm) * MDIM + k + half * 2 + 0];
        a.y = sD[(mtR + lm) * MDIM + k + half * 2 + 1];
        b.x = sZ[(k + half * 2 + 0) * NB + ntR + lm];
        b.y = sZ[(k + half * 2 + 1) * NB + ntR + lm];
        acc = wmma4(a, b, acc);
      }
#pragma unroll
      for (int r = 0; r < 8; ++r) {
        const int m = mtR + r + 8 * half, n = ntR + lm;
        sR[m * NB + n] = acc[r] - sX[m * NB + n];
      }
    }
    __syncthreads();

    // ---- phase 2: z = shrink(z - eta * D^T res) ----
    float dsq = 0.f;
#pragma unroll
    for (int tz = 0; tz < 4; ++tz) {
      const int mb = (wid + 8 * (tz >> 1)) * 16;
      const int nb = (tz & 1) * 16;
      v8f acc = {0.f, 0.f, 0.f, 0.f, 0.f, 0.f, 0.f, 0.f};
      for (int k = 0; k < NDIM; k += 4) {
        v2f a, b;
        a.x = sD[(k + half * 2 + 0) * MDIM + mb + lm];  // (D^T)[m][k]
        a.y = sD[(k + half * 2 + 1) * MDIM + mb + lm];
        b.x = sR[(k + half * 2 + 0) * NB + nb + lm];
        b.y = sR[(k + half * 2 + 1) * NB + nb + lm];
        acc = wmma4(a, b, acc);
      }
#pragma unroll
      for (int r = 0; r < 8; ++r) {
        const int m = mb + r + 8 * half, n = nb + lm;
        const float zo = sZ[m * NB + n];
        const float v  = zo - eta * acc[r];
        const float av = fabsf(v) - thr;
        const float zn = av > 0.f ? copysignf(av, v) : 0.f;
        const float d  = zn - zo;
        dsq += d * d;
        sZ[m * NB + n] = zn;
      }
    }
    atomicAdd(&sDelta, dsq);
    __syncthreads();
    const float tot = sDelta;
    __syncthreads();
    if (tid == 0) sDelta = 0.f;     // ordered before next accumulation by phase-1 barrier
    if (tot == 0.f) break;          // uniform across block
  }
  __syncthreads();

  for (int i = tid; i < MDIM * NB; i += 256) {
    const int m = i >> 5, j = i & 31;
    zws[m * N_SAMP + n0 + j] = sZ[i];
  }
}

// ---------------------------------------------------------------------------
// Kernel 3: per-sample support refinement. One 64-thread block per sample.
//   Solve G_SS a = D_S^T x_i - lambd*sign_S via Cholesky in LDS, y = w_S . a
// ---------------------------------------------------------------------------
__global__ void k_refine(const float* __restrict__ x, const float* __restrict__ D,
                         const float* __restrict__ w, const float* __restrict__ G,
                         const float* __restrict__ zws, float* __restrict__ out) {
  __shared__ float sA[SMAX * SMAX];   // ~36 KB
  __shared__ float sB[SMAX];
  __shared__ float sSol[SMAX];
  __shared__ float sSgn[SMAX];
  __shared__ int   sIdx[SMAX];
  __shared__ int   sCnt;

  const int i = blockIdx.x;
  const int t = threadIdx.x;          // 64 threads

  if (t == 0) {
    int c = 0;
    for (int j = 0; j < MDIM; ++j) {
      const float zj = zws[j * N_SAMP + i];
      if (zj != 0.f && c < SMAX) {
        sIdx[c] = j;
        sSgn[c] = zj > 0.f ? 1.f : -1.f;
        ++c;
      }
    }
    sCnt = c;
  }
  __syncthreads();
  const int s = sCnt;

  for (int e = t; e < s * s; e += 64) {
    const int r = e / s, c = e % s;
    sA[r * SMAX + c] = G[sIdx[r] * MDIM + sIdx[c]];
  }
  for (int r = t; r < s; r += 64) {
    const int j = sIdx[r];
    float dot = 0.f;
    for (int k = 0; k < NDIM; ++k) dot += D[k * MDIM + j] * x[i * NDIM + k];
    sB[r] = dot - LAMBD * sSgn[r];
  }
  __syncthreads();

  // Cholesky (lower, right-looking) on s x s SPD Gram submatrix
  for (int k = 0; k < s; ++k) {
    if (t == 0) sA[k * SMAX + k] = sqrtf(fmaxf(sA[k * SMAX + k], 1e-20f));
    __syncthreads();
    const float dk = sA[k * SMAX + k];
    for (int r = k + 1 + t; r < s; r += 64) sA[r * SMAX + k] /= dk;
    __syncthreads();
    for (int r = k + 1 + t; r < s; r += 64) {
      const float lrk = sA[r * SMAX + k];
      for (int c = k + 1; c <= r; ++c) sA[r * SMAX + c] -= lrk * sA[c * SMAX + k];
    }
    __syncthreads();
  }

  if (t == 0) {
    for (int k = 0; k < s; ++k) {         // L y = b
      float acc = sB[k];
      for (int j = 0; j < k; ++j) acc -= sA[k * SMAX + j] * sSol[j];
      sSol[k] = acc / sA[k * SMAX + k];
    }
    for (int k = s - 1; k >= 0; --k) {    // L^T a = y (into sB)
      float acc = sSol[k];
      for (int j = k + 1; j < s; ++j) acc -= sA[j * SMAX + k] * sB[j];
      sB[k] = acc / sA[k * SMAX + k];
    }
    float y = 0.f;
    for (int r = 0; r < s; ++r) y += w[sIdx[r]] * sB[r];
    out[i * 2 + 0] = y;
    out[i * 2 + 1] = -y;
  }
}

// ---------------------------------------------------------------------------
extern "C" void kernel_launch(void* const* d_in, const int* in_sizes, int n_in,
                              void* d_out, int out_size, void* d_ws, size_t ws_size,
                              hipStream_t stream) {
  (void)in_sizes; (void)n_in; (void)out_size; (void)ws_size;
  const float* x = (const float*)d_in[0];   // (1024, 64)
  const float* D = (const float*)d_in[1];   // (64, 256)
  const float* w = (const float*)d_in[2];   // (256,)
  float* out = (float*)d_out;               // (1024, 2)

  float* ws   = (float*)d_ws;
  float* etaW = ws;                         // 64 floats reserved
  float* G    = ws + 64;                    // 256*256 floats
  float* zws  = G + MDIM * MDIM;            // 256*1024 floats

  hipLaunchKernelGGL(k_gram_eta, dim3(1),            dim3(256), 0, stream, D, G, etaW);
  hipLaunchKernelGGL(k_ista,     dim3(N_SAMP / NB),  dim3(256), 0, stream, x, D, etaW, zws);
  hipLaunchKernelGGL(k_refine,   dim3(N_SAMP),       dim3(64),  0, stream, x, D, w, G, zws, out);
}